// HexagonalSensor_83133386982139
// MI455X (gfx1250) — compile-verified
//
#include <hip/hip_runtime.h>
#include <math.h>

#ifndef M_PI
#define M_PI 3.14159265358979323846
#endif

#define RINGR 25
#define NPIX  1951

typedef float v4f __attribute__((ext_vector_type(4)));   // native clang vector:
                                                         // valid for nontemporal builtins

struct HexConsts {
  float offx, offy;   // grid offset (center nearest origin)
  float c, s;         // cos(-rot), sin(-rot), f32
  float s33;          // f32(1/sqrt(3))
  float hs;           // f32 HEX_SIZE
  float c1;           // f32(HEX_SIZE*sqrt(3))
  float c2;           // f32(HEX_SIZE*1.5)
  float s32;          // f32(sqrt(3)/2)
  float hir;          // f32(HEX_SIZE*sqrt(3)/2) = inradius
  float thr;          // f32(1 - EDGE_WIDTH/inradius)
  int   k;            // 0: identity axial frame, 1: rot ~ 60 deg (hex symmetry)
};

__device__ __forceinline__ void hex_accum(float x, float y, float val,
                                          const HexConsts& hc,
                                          float* __restrict__ hist)
{
#pragma clang fp contract(off)
  // world -> grid frame (matches XLA: separate mul/sub, no fma)
  float x0 = x - hc.offx;
  float y0 = y - hc.offy;
  float xg = hc.c * x0 - hc.s * y0;
  float yg = hc.s * x0 + hc.c * y0;
  // fractional axial coords (IEEE divides to match reference; /1.0f is identity)
  float q = hc.s33 * xg - yg / 3.0f;
  float r = 2.0f * yg / 3.0f;
  if (hc.hs != 1.0f) { q = q / hc.hs; r = r / hc.hs; }
  // cube rounding (round-half-even == jnp.round)
  float ss = -q - r;
  float qi = rintf(q), ri = rintf(r), si = rintf(ss);
  float dq = fabsf(qi - q), dr = fabsf(ri - r), dsv = fabsf(si - ss);
  if (dq > dr && dq > dsv)      qi = -ri - si;   // mutually exclusive with next,
  else if (dr > dq && dr > dsv) ri = -qi - si;   // matches sequential updates
  // edge rejection via hex norm (in rotated frame, like reference)
  float hcx = hc.c1 * (qi + ri / 2.0f);
  float hcy = hc.c2 * ri;
  float dxg = xg - hcx;
  float dyg = yg - hcy;
  float ax = fabsf(dxg), ay = fabsf(dyg);
  float hd = fmaxf(ay, hc.s32 * ax + 0.5f * ay) / hc.hir;
  if (hd > hc.thr) return;
  // inverse of the lookup table, analytically.
  int qa = (int)qi, ra = (int)ri;
  int qb = hc.k ? (-ra)      : qa;   // A^-1 for a -60deg table rotation
  int rb = hc.k ? (qa + ra)  : ra;
  int sb = qb + rb;
  if (qb < -RINGR || qb > RINGR || rb < -RINGR || rb > RINGR ||
      sb < -RINGR || sb > RINGR) return;          // == rect-bounds & tab>=0
  int idx;
  if (qb <= 0) {
    int nn = 50 + qb;                              // offset = S(50+q) - S(25)
    idx = nn * (nn + 1) / 2 - 325 + (rb + qb + RINGR);
  } else {
    idx = 950 + 51 * qb - (qb * (qb - 1)) / 2 + (rb + RINGR);
  }
  atomicAdd(&hist[idx], val);                      // ds_add_f32 (LDS, wave32)
}

__global__ void __launch_bounds__(256)
hex_hist_kernel(const v4f* __restrict__ x4, const v4f* __restrict__ y4,
                const v4f* __restrict__ v4,
                const float* __restrict__ xs, const float* __restrict__ ys,
                const float* __restrict__ vs,
                float* __restrict__ out, int n4, int n, HexConsts hc)
{
  __shared__ float hist[NPIX];
  for (int i = threadIdx.x; i < NPIX; i += blockDim.x) hist[i] = 0.0f;
  __syncthreads();

  int tid    = blockIdx.x * blockDim.x + threadIdx.x;
  int stride = gridDim.x * blockDim.x;
  for (int i = tid; i < n4; i += stride) {
    if (i + stride < n4) {                 // stage next chunk into L2 (192 MB)
      __builtin_prefetch(&x4[i + stride]); // global_prefetch_b8
      __builtin_prefetch(&y4[i + stride]);
      __builtin_prefetch(&v4[i + stride]);
    }
    // read-once streams: non-temporal B128 loads
    v4f xv = __builtin_nontemporal_load(&x4[i]);
    v4f yv = __builtin_nontemporal_load(&y4[i]);
    v4f vv = __builtin_nontemporal_load(&v4[i]);
    hex_accum(xv.x, yv.x, vv.x, hc, hist);
    hex_accum(xv.y, yv.y, vv.y, hc, hist);
    hex_accum(xv.z, yv.z, vv.z, hc, hist);
    hex_accum(xv.w, yv.w, vv.w, hc, hist);
  }
  if (blockIdx.x == 0 && threadIdx.x == 0) {       // scalar tail (n % 4)
    for (int i = 4 * n4; i < n; ++i) hex_accum(xs[i], ys[i], vs[i], hc, hist);
  }
  __syncthreads();
  for (int i = threadIdx.x; i < NPIX; i += blockDim.x) {
    unsafeAtomicAdd(&out[i], hist[i]);             // global_atomic_add_f32
  }
}

__global__ void zero_kernel(float* __restrict__ out, int n) {
  int i = blockIdx.x * blockDim.x + threadIdx.x;
  if (i < n) out[i] = 0.0f;
}

// ---------------- host: replicate numpy init (float32 semantics) -------------
static void build_centers(float* cx, float* cy) {
  int n = 0;
  for (int q = -RINGR; q <= RINGR; ++q) {
    int rlo = (-q - RINGR) > -RINGR ? (-q - RINGR) : -RINGR;
    int rhi = (-q + RINGR) < RINGR ? (-q + RINGR) : RINGR;
    for (int r = rlo; r <= rhi && n < NPIX; ++r) {
      cx[n] = (float)(1.7320508075688772 * ((double)q + (double)r / 2.0));
      cy[n] = (float)(1.5 * (double)r);
      ++n;
    }
  }
}

extern "C" void kernel_launch(void* const* d_in, const int* in_sizes, int n_in,
                              void* d_out, int out_size, void* d_ws, size_t ws_size,
                              hipStream_t stream) {
  (void)n_in; (void)d_ws; (void)ws_size;
  const float* dx = (const float*)d_in[0];
  const float* dy = (const float*)d_in[1];
  const float* dv = (const float*)d_in[2];
  float* out = (float*)d_out;
  int n = in_sizes[0];

  // --- _detect_hex_grid in float32, matching numpy (first-occurrence argmin) ---
  static float cxs[NPIX], cys[NPIX];   // recomputed every call; pure function
  build_centers(cxs, cys);
  float mind2 = INFINITY; int bi = 0, bj = 0;
  for (int i = 0; i < NPIX; ++i)
    for (int j = 0; j < NPIX; ++j) {
      if (i == j) continue;
      float ddx = cxs[i] - cxs[j];
      float ddy = cys[i] - cys[j];
      float d2  = ddx * ddx + ddy * ddy;     // f32, same order as (d**2).sum(-1)
      if (d2 < mind2) { mind2 = d2; bi = i; bj = j; }
    }
  float min_dist = sqrtf(mind2);
  float hex_size = min_dist / (float)1.7320508075688772;   // NEP50: f32 op
  float vx = cxs[bi] - cxs[bj];
  float vy = cys[bi] - cys[bj];
  float ang = (float)atan2((double)vy, (double)vx);
  float p3  = (float)(M_PI / 3.0);
  float rot = fmodf(ang, p3);
  if (rot != 0.0f && rot < 0.0f) rot += p3;                // np.mod semantics
  float bestn = INFINITY; int oi = 0;
  for (int i = 0; i < NPIX; ++i) {
    float nn = cxs[i] * cxs[i] + cys[i] * cys[i];
    if (nn < bestn) { bestn = nn; oi = i; }
  }

  double rot_d = (double)rot;
  double hs_d  = (double)hex_size;
  HexConsts hc;
  hc.offx = cxs[oi];
  hc.offy = cys[oi];
  hc.c    = (float)cos(-rot_d);
  hc.s    = (float)sin(-rot_d);
  hc.s33  = (float)0.5773502691896257;
  hc.hs   = hex_size;
  hc.c1   = (float)(hs_d * 1.7320508075688772);            // folded f64 -> f32
  hc.c2   = (float)(hs_d * 1.5);
  hc.s32  = (float)0.8660254037844386;
  hc.hir  = (float)(hs_d * 0.8660254037844386);
  hc.thr  = (float)(1.0 - 0.1 / (hs_d * 0.8660254037844386));
  hc.k    = (rot_d > M_PI / 6.0) ? 1 : 0;                  // 0 or ~pi/3 only

  zero_kernel<<<(out_size + 255) / 256, 256, 0, stream>>>(out, out_size);

  int n4 = n >> 2;
  const int NB = 1024, NT = 256;                           // 8 wave32 per block
  hex_hist_kernel<<<NB, NT, 0, stream>>>(
      (const v4f*)dx, (const v4f*)dy, (const v4f*)dv,
      dx, dy, dv, out, n4, n, hc);
}